// FRC_Loss_59038620450991
// MI455X (gfx1250) — compile-verified
//
#include <hip/hip_runtime.h>
#include <math.h>

typedef float v2f __attribute__((ext_vector_type(2)));
typedef float v8f __attribute__((ext_vector_type(8)));

#define FFTN   512
#define PLANE  (FFTN * FFTN)        // 262144 floats per image plane
#define RNY    256                  // Nyquist ring
#define NBINS  258                  // rings 0..256 + overflow ring 257
#define NBATCH 32

// workspace layout (in floats)
#define OFF_WC   0                        // cos DFT matrix  [512][512]
#define OFF_WS   (PLANE)                  // -sin DFT matrix [512][512]
#define OFF_T    (2 * PLANE)              // T[img(2)][re/im(2)][512][512]
#define OFF_BINS (6 * PLANE)              // bins[258][32][4]
#define BINS_FLOATS (NBINS * NBATCH * 4)

// ---------------------------------------------------------------------------
// Build W = exp(-2*pi*i*k*j/512)/512 (forward-normalized per 1-D stage, so the
// two stages give the 1/(H*W) of norm='forward'). Phase reduced mod 512 so the
// angle stays in [-2pi, 0] exactly.
// ---------------------------------------------------------------------------
__global__ void build_dft(float* __restrict__ ws) {
    int t = blockIdx.x * blockDim.x + threadIdx.x;
    if (t >= PLANE) return;
    int k = t >> 9;
    int j = t & (FFTN - 1);
    int m = (k * j) & (FFTN - 1);
    float ang = -3.14159265358979323846f * (float)m * (1.0f / 256.0f); // -2*pi*m/512
    float s, c;
    __sincosf(ang, &s, &c);
    ws[OFF_WC + t] = c * (1.0f / 512.0f);
    ws[OFF_WS + t] = s * (1.0f / 512.0f);
}

__global__ void zero_bins(float* __restrict__ ws) {
    int t = blockIdx.x * blockDim.x + threadIdx.x;
    if (t < BINS_FLOATS) ws[OFF_BINS + t] = 0.0f;
}

// ---------------------------------------------------------------------------
// Stage 1: T = X * W  (row-direction FFT).  One wave per 16x16 output tile of
// one image; two f32 WMMA accumulators (real & imag) over K=512 in steps of 4.
// A-frag (16x4 f32): lane = M (0..15 in each half), VGPR pair = {K+kh, K+kh+1},
// kh = 2*(lane>=16).  B-frag (4x16): lane = N, VGPR pair = rows {K+kh, K+kh+1}.
// ---------------------------------------------------------------------------
__global__ void row_fft(const float* __restrict__ outp,
                        const float* __restrict__ targ,
                        float* __restrict__ ws, int b) {
    const int lane  = threadIdx.x & 31;
    const int wave  = threadIdx.x >> 5;
    const int tile  = blockIdx.x * (blockDim.x >> 5) + wave;  // 0..2047
    const int img   = tile >> 10;                             // 0=output 1=target
    const int ib    = (tile >> 5) & 31;
    const int jb    = tile & 31;

    const float* X   = (img ? targ : outp) + (size_t)b * PLANE;
    const float* Wc  = ws + OFF_WC;
    const float* Wsn = ws + OFF_WS;

    const int mrow  = ib * 16 + (lane & 15);
    const int khalf = (lane >> 4) << 1;            // 0 or 2
    const int ncol  = jb * 16 + (lane & 15);

    v8f accr = {};
    v8f acci = {};
    for (int k = 0; k < FFTN; k += 4) {
        const int kb = k + khalf;
        v2f a = *(const v2f*)(X + mrow * FFTN + kb);           // 8B aligned
        v2f bc, bs;
        bc.x = Wc [(kb + 0) * FFTN + ncol];
        bc.y = Wc [(kb + 1) * FFTN + ncol];
        bs.x = Wsn[(kb + 0) * FFTN + ncol];
        bs.y = Wsn[(kb + 1) * FFTN + ncol];
        accr = __builtin_amdgcn_wmma_f32_16x16x4_f32(false, a, false, bc,
                                                     (short)0, accr, false, false);
        acci = __builtin_amdgcn_wmma_f32_16x16x4_f32(false, a, false, bs,
                                                     (short)0, acci, false, false);
    }

    float* Tr = ws + OFF_T + (size_t)img * 2 * PLANE;
    float* Ti = Tr + PLANE;
    const int rbase = ib * 16 + 8 * (lane >> 4);
#pragma unroll
    for (int v = 0; v < 8; ++v) {
        const int r = rbase + v;
        Tr[r * FFTN + ncol] = accr[v];
        Ti[r * FFTN + ncol] = acci[v];
    }
}

// ---------------------------------------------------------------------------
// Stage 2: F = W * T (column FFT) for BOTH images of this batch, fused with the
// spectral products + radial binning.  Fr = Wc*Tr - Ws*Ti ; Fi = Wc*Ti + Ws*Tr.
// 8 f32 WMMA accumulators (4 per image); F tiles never hit memory.
// ---------------------------------------------------------------------------
__global__ void col_fft_reduce(float* __restrict__ ws, int b) {
    const int lane = threadIdx.x & 31;
    const int wave = threadIdx.x >> 5;
    const int tile = blockIdx.x * (blockDim.x >> 5) + wave;   // 0..1023
    const int ub   = tile >> 5;
    const int vb   = tile & 31;

    const float* Wc  = ws + OFF_WC;
    const float* Wsn = ws + OFF_WS;
    const float* Tr0 = ws + OFF_T;
    const float* Ti0 = Tr0 + PLANE;
    const float* Tr1 = Tr0 + 2 * PLANE;
    const float* Ti1 = Tr1 + PLANE;

    const int arow  = ub * 16 + (lane & 15);
    const int khalf = (lane >> 4) << 1;
    const int ncol  = vb * 16 + (lane & 15);

    v8f P0 = {}, Q0 = {}, R0 = {}, S0 = {};
    v8f P1 = {}, Q1 = {}, R1 = {}, S1 = {};
    for (int k = 0; k < FFTN; k += 4) {
        const int kb = k + khalf;
        v2f ac = *(const v2f*)(Wc  + arow * FFTN + kb);
        v2f as = *(const v2f*)(Wsn + arow * FFTN + kb);
        v2f br0, bi0, br1, bi1;
        br0.x = Tr0[(kb + 0) * FFTN + ncol]; br0.y = Tr0[(kb + 1) * FFTN + ncol];
        bi0.x = Ti0[(kb + 0) * FFTN + ncol]; bi0.y = Ti0[(kb + 1) * FFTN + ncol];
        br1.x = Tr1[(kb + 0) * FFTN + ncol]; br1.y = Tr1[(kb + 1) * FFTN + ncol];
        bi1.x = Ti1[(kb + 0) * FFTN + ncol]; bi1.y = Ti1[(kb + 1) * FFTN + ncol];
        P0 = __builtin_amdgcn_wmma_f32_16x16x4_f32(false, ac, false, br0, (short)0, P0, false, false);
        Q0 = __builtin_amdgcn_wmma_f32_16x16x4_f32(false, as, false, bi0, (short)0, Q0, false, false);
        R0 = __builtin_amdgcn_wmma_f32_16x16x4_f32(false, ac, false, bi0, (short)0, R0, false, false);
        S0 = __builtin_amdgcn_wmma_f32_16x16x4_f32(false, as, false, br0, (short)0, S0, false, false);
        P1 = __builtin_amdgcn_wmma_f32_16x16x4_f32(false, ac, false, br1, (short)0, P1, false, false);
        Q1 = __builtin_amdgcn_wmma_f32_16x16x4_f32(false, as, false, bi1, (short)0, Q1, false, false);
        R1 = __builtin_amdgcn_wmma_f32_16x16x4_f32(false, ac, false, bi1, (short)0, R1, false, false);
        S1 = __builtin_amdgcn_wmma_f32_16x16x4_f32(false, as, false, br1, (short)0, S1, false, false);
    }

    float* bins = ws + OFF_BINS;
#pragma unroll
    for (int v = 0; v < 8; ++v) {
        const int u  = ub * 16 + v + 8 * (lane >> 4);   // C/D layout: M = v (+8 upper half)
        const int vv = ncol;
        const float fx = (u  < 256) ? (float)u  : (float)(u  - 512);
        const float fy = (vv < 256) ? (float)vv : (float)(vv - 512);
        int r = (int)rintf(sqrtf(fx * fx + fy * fy));
        if (r > RNY) r = RNY + 1;

        const float f1r = P0[v] - Q0[v];
        const float f1i = R0[v] + S0[v];
        const float f2r = P1[v] - Q1[v];
        const float f2i = R1[v] + S1[v];

        const float av = f1r * f2r + f1i * f2i;
        const float bv = f1i * f2r - f1r * f2i;
        const float c1 = f1r * f1r + f1i * f1i;
        const float c2 = f2r * f2r + f2i * f2i;

        float* base = bins + ((size_t)r * NBATCH + b) * 4;
        unsafeAtomicAdd(base + 0, av);   // global_atomic_add_f32
        unsafeAtomicAdd(base + 1, bv);
        unsafeAtomicAdd(base + 2, c1);
        unsafeAtomicAdd(base + 3, c2);
    }
}

// ---------------------------------------------------------------------------
// Final reduction: FRC per (ring, batch), then mean((1-frc)^2) over 257*32.
// ---------------------------------------------------------------------------
__global__ void finalize(const float* __restrict__ ws, float* __restrict__ out) {
    __shared__ float red[256];
    const float* bins = ws + OFF_BINS;
    const int total = (RNY + 1) * NBATCH;   // 8224
    float local = 0.0f;
    for (int i = threadIdx.x; i < total; i += blockDim.x) {
        const float* s = bins + (size_t)i * 4;
        const float Cr = s[0], Ci = s[1], C1 = s[2], C2 = s[3];
        const float frc = sqrtf(Cr * Cr + Ci * Ci) / (sqrtf(C1 * C2) + 1e-8f);
        const float d = 1.0f - frc;
        local += d * d;
    }
    red[threadIdx.x] = local;
    __syncthreads();
    for (int s = 128; s > 0; s >>= 1) {
        if ((int)threadIdx.x < s) red[threadIdx.x] += red[threadIdx.x + s];
        __syncthreads();
    }
    if (threadIdx.x == 0) out[0] = red[0] / (float)total;
}

extern "C" void kernel_launch(void* const* d_in, const int* in_sizes, int n_in,
                              void* d_out, int out_size, void* d_ws, size_t ws_size,
                              hipStream_t stream) {
    const float* outp = (const float*)d_in[0];
    const float* targ = (const float*)d_in[1];
    float* ws  = (float*)d_ws;
    float* out = (float*)d_out;

    build_dft<<<PLANE / 256, 256, 0, stream>>>(ws);
    zero_bins<<<(BINS_FLOATS + 255) / 256, 256, 0, stream>>>(ws);

    for (int b = 0; b < NBATCH; ++b) {
        // 2 images * 32*32 tiles = 2048 tiles, 8 waves (tiles) per 256-thread block
        row_fft<<<2048 / 8, 256, 0, stream>>>(outp, targ, ws, b);
        // 32*32 tiles over (u,v), both images handled inside
        col_fft_reduce<<<1024 / 8, 256, 0, stream>>>(ws, b);
    }

    finalize<<<1, 256, 0, stream>>>(ws, out);
}